// Normal_GCN_53884659695767
// MI455X (gfx1250) — compile-verified
//
#include <hip/hip_runtime.h>
#include <math.h>

typedef __attribute__((ext_vector_type(2))) float v2f;
typedef __attribute__((ext_vector_type(8))) float v8f;

// ---------------- degree / dinv ----------------
__global__ void fill_ones_kernel(float* p, int n) {
    int i = blockIdx.x * blockDim.x + threadIdx.x;
    if (i < n) p[i] = 1.0f;
}

__global__ void count_deg_kernel(const int* __restrict__ dst, float* __restrict__ deg, int E) {
    int e = blockIdx.x * blockDim.x + threadIdx.x;
    if (e < E) atomicAdd(&deg[dst[e]], 1.0f);
}

__global__ void rsqrt_inplace_kernel(float* p, int n) {
    int i = blockIdx.x * blockDim.x + threadIdx.x;
    if (i < n) p[i] = rsqrtf(p[i]);
}

// ---------------- GEMM via V_WMMA_F32_16X16X4_F32 ----------------
// Out[M x Ncols] = A[M x K] * W[K x Ncols], row-major, M multiple of 16,
// K multiple of 4, Ncols multiple of 16. One wave computes one 16x16 tile;
// wave w of a block handles column tile w; blockIdx.x is the row tile.
__global__ __launch_bounds__(256) void gemm_wmma_f32(
    const float* __restrict__ A, const float* __restrict__ W,
    float* __restrict__ Out, int K, int Ncols) {
    const int wave = threadIdx.x >> 5;
    const int lane = threadIdx.x & 31;
    const int half = lane >> 4;     // 0: lanes 0-15, 1: lanes 16-31
    const int l16  = lane & 15;
    const int row0 = blockIdx.x * 16;
    const int col0 = wave * 16;

    // A fragment: lane<16 -> row=l16, K pair {k,k+1}; lane>=16 -> K pair {k+2,k+3}
    const float* arow = A + (size_t)(row0 + l16) * K + half * 2;
    // B fragment: lane<16 -> rows {k,k+1} of W at col l16; lane>=16 -> rows {k+2,k+3}
    const float* wcol = W + (size_t)(half * 2) * Ncols + col0 + l16;

    v8f c = {};
    for (int k = 0; k < K; k += 4) {
        v2f a, b;
        a.x = arow[k];
        a.y = arow[k + 1];
        b.x = wcol[(size_t)k * Ncols];
        b.y = wcol[(size_t)(k + 1) * Ncols];
        c = __builtin_amdgcn_wmma_f32_16x16x4_f32(
                /*neg_a=*/false, a, /*neg_b=*/false, b,
                /*c_mod=*/(short)0, c, /*reuse_a=*/false, /*reuse_b=*/false);
    }

    // C/D layout: VGPR r, lanes 0-15 -> M=r, N=lane; lanes 16-31 -> M=r+8, N=lane-16
    float* orow = Out + (size_t)(row0 + half * 8) * Ncols + col0 + l16;
#pragma unroll
    for (int r = 0; r < 8; ++r)
        orow[(size_t)r * Ncols] = c[r];
}

// ---------------- aggregation helpers ----------------
// agg[n][f] = t[n][f] * dinv[n]^2 + b[f]     (self-loop + bias init)
__global__ void init_agg_kernel(const float* __restrict__ t, const float* __restrict__ dinv,
                                const float* __restrict__ b, float* __restrict__ agg,
                                int shiftF, long long total) {
    long long idx = (long long)blockIdx.x * blockDim.x + threadIdx.x;
    if (idx >= total) return;
    int n = (int)(idx >> shiftF);
    int f = (int)(idx & ((1 << shiftF) - 1));
    float di = dinv[n];
    agg[idx] = t[idx] * di * di + b[f];
}

// agg[dst][:] += t[src][:] * dinv[src]*dinv[dst], one wave32 per edge.
// Edge index is forced wave-uniform so src/dst/dinv loads scalarize (one
// SMEM fetch per wave instead of 32 redundant per-lane VMEM fetches). Each lane moves
// F/32 contiguous floats with a single wide load, then issues f32 atomics.
template <int F>
__global__ __launch_bounds__(256) void scatter_edges_wave(
    const float* __restrict__ t, const int* __restrict__ src,
    const int* __restrict__ dst, const float* __restrict__ dinv,
    float* __restrict__ agg, int E) {
    int e = blockIdx.x * (blockDim.x >> 5) + (threadIdx.x >> 5);
    if (e >= E) return;                       // wave-uniform guard
    e = __builtin_amdgcn_readfirstlane(e);    // force SGPR -> scalar loads
    const int s = src[e];
    const int d = dst[e];
    const float norm = dinv[s] * dinv[d];
    const int lane = threadIdx.x & 31;
    constexpr int VPT = F / 32;               // 4 (F=128) or 2 (F=64)
    const float* hp = t + (size_t)s * F + lane * VPT;
    float* ap = agg + (size_t)d * F + lane * VPT;
    if constexpr (VPT == 4) {
        float4 v = *(const float4*)hp;        // global_load_b128
        atomicAdd(&ap[0], v.x * norm);
        atomicAdd(&ap[1], v.y * norm);
        atomicAdd(&ap[2], v.z * norm);
        atomicAdd(&ap[3], v.w * norm);
    } else {
        float2 v = *(const float2*)hp;        // global_load_b64
        atomicAdd(&ap[0], v.x * norm);
        atomicAdd(&ap[1], v.y * norm);
    }
}

__global__ void relu_kernel(float* p, long long n) {
    long long i = (long long)blockIdx.x * blockDim.x + threadIdx.x;
    if (i < n) p[i] = fmaxf(p[i], 0.0f);
}

// ---------------- log_softmax over 64 cols, one wave32 per row ----------------
__global__ void log_softmax64_kernel(const float* __restrict__ in, float* __restrict__ out, int N) {
    int row = blockIdx.x * (blockDim.x >> 5) + (threadIdx.x >> 5);
    int lane = threadIdx.x & 31;
    if (row >= N) return;
    const float* r = in + (size_t)row * 64;
    float v0 = r[lane];
    float v1 = r[lane + 32];
    float m = fmaxf(v0, v1);
#pragma unroll
    for (int o = 16; o > 0; o >>= 1) m = fmaxf(m, __shfl_xor(m, o, 32));
    float s = __expf(v0 - m) + __expf(v1 - m);
#pragma unroll
    for (int o = 16; o > 0; o >>= 1) s += __shfl_xor(s, o, 32);
    float lse = m + __logf(s);
    out[(size_t)row * 64 + lane] = v0 - lse;
    out[(size_t)row * 64 + lane + 32] = v1 - lse;
}

// ---------------- driver ----------------
extern "C" void kernel_launch(void* const* d_in, const int* in_sizes, int n_in,
                              void* d_out, int out_size, void* d_ws, size_t ws_size,
                              hipStream_t stream) {
    const float* x  = (const float*)d_in[0];
    const int*   ei = (const int*)  d_in[1];
    const float* W0 = (const float*)d_in[2];
    const float* b0 = (const float*)d_in[3];
    const float* W1 = (const float*)d_in[4];
    const float* b1 = (const float*)d_in[5];
    const float* W2 = (const float*)d_in[6];
    const float* b2 = (const float*)d_in[7];

    const int DIN = 256, DH = 128, DOUT = 64;
    const int N = in_sizes[0] / DIN;   // 50000 (multiple of 16)
    const int E = in_sizes[1] / 2;     // 800000
    const int* src = ei;
    const int* dst = ei + E;

    char* ws = (char*)d_ws;
    size_t offA = ((size_t)N * sizeof(float) + 255) & ~(size_t)255;
    size_t offB = offA + (size_t)N * DH * sizeof(float);
    float* dinv = (float*)ws;
    float* bufA = (float*)(ws + offA);   // GEMM output / scatter source
    float* bufB = (float*)(ws + offB);   // aggregation target / next-layer input

    const int TB = 256;
    auto blocks1 = [&](long long n) { return (unsigned)((n + TB - 1) / TB); };
    const unsigned edgeBlocks = (unsigned)((E + 7) / 8);   // 8 waves per block

    // degrees -> dinv
    fill_ones_kernel<<<blocks1(N), TB, 0, stream>>>(dinv, N);
    count_deg_kernel<<<blocks1(E), TB, 0, stream>>>(dst, dinv, E);
    rsqrt_inplace_kernel<<<blocks1(N), TB, 0, stream>>>(dinv, N);

    const int rowTiles = N / 16;
    long long tot128 = (long long)N * DH;
    long long tot64  = (long long)N * DOUT;

    // ---- layer 0: x[N,256] @ W0[256,128] ----
    gemm_wmma_f32<<<rowTiles, 32 * (DH / 16), 0, stream>>>(x, W0, bufA, DIN, DH);
    init_agg_kernel<<<blocks1(tot128), TB, 0, stream>>>(bufA, dinv, b0, bufB, 7, tot128);
    scatter_edges_wave<128><<<edgeBlocks, TB, 0, stream>>>(bufA, src, dst, dinv, bufB, E);
    relu_kernel<<<blocks1(tot128), TB, 0, stream>>>(bufB, tot128);

    // ---- layer 1: h[N,128] @ W1[128,128] ----
    gemm_wmma_f32<<<rowTiles, 32 * (DH / 16), 0, stream>>>(bufB, W1, bufA, DH, DH);
    init_agg_kernel<<<blocks1(tot128), TB, 0, stream>>>(bufA, dinv, b1, bufB, 7, tot128);
    scatter_edges_wave<128><<<edgeBlocks, TB, 0, stream>>>(bufA, src, dst, dinv, bufB, E);
    relu_kernel<<<blocks1(tot128), TB, 0, stream>>>(bufB, tot128);

    // ---- layer 2: h[N,128] @ W2[128,64] ----
    gemm_wmma_f32<<<rowTiles, 32 * (DOUT / 16), 0, stream>>>(bufB, W2, bufA, DH, DOUT);
    init_agg_kernel<<<blocks1(tot64), TB, 0, stream>>>(bufA, dinv, b2, bufB, 6, tot64);
    scatter_edges_wave<64><<<edgeBlocks, TB, 0, stream>>>(bufA, src, dst, dinv, bufB, E);

    // ---- log_softmax -> d_out ----
    log_softmax64_kernel<<<(N + 7) / 8, 256, 0, stream>>>(bufB, (float*)d_out, N);
}